// YOLOv2Loss_33200097198232
// MI455X (gfx1250) — compile-verified
//
#include <hip/hip_runtime.h>
#include <math.h>

// ---------------- problem constants ----------------
#define QN      128
#define SSN     361            // 19*19
#define NA      5
#define ND      85
#define NC      80
#define CELLS   (QN * SSN)     // 46208
#define CPB     8              // cells per block (one wave32 per cell)
#define NBLK    (CELLS / CPB)  // 5776 (exact)
#define TPB     256            // 8 waves
#define CELL_F  (NA * ND)      // 425 floats per cell
#define BLK_F   (CPB * CELL_F) // 3400 floats per block slab
#define BLK_V4  (BLK_F / 4)    // 850 x 16B chunks (13600 B, 16B aligned per block)

// ---------------- CDNA5 async global->LDS path ----------------
#if defined(__has_builtin)
#if __has_builtin(__builtin_amdgcn_global_load_async_to_lds_b128)
#define HAVE_ASYNC_LDS 1
#endif
#if __has_builtin(__builtin_amdgcn_s_wait_asynccnt)
#define HAVE_WAIT_ASYNC 1
#endif
#endif

typedef int v4i_t __attribute__((vector_size(4 * sizeof(int))));
typedef __attribute__((address_space(1))) v4i_t* gbl_v4i_p;
typedef __attribute__((address_space(3))) v4i_t* lds_v4i_p;

__device__ __forceinline__ float softplus_(float x) {
  // jax.nn.softplus == logaddexp(x, 0) == max(x,0) + log1p(exp(-|x|))
  return fmaxf(x, 0.f) + log1pf(expf(-fabsf(x)));
}
__device__ __forceinline__ float sigmoid_(float x) {
  return 1.f / (1.f + expf(-x));
}

__global__ __launch_bounds__(TPB) void yolo_loss_partial(
    const float* __restrict__ pred, const float* __restrict__ targ,
    float* __restrict__ partials) {
  __shared__ float sP[BLK_F];
  __shared__ float sT[BLK_F];
  __shared__ float sBox[CPB][40];   // [0..19] decoded pred boxes, [20..39] gt boxes
  __shared__ float sIou[CPB][25];   // [ag*5 + ap]
  __shared__ int   sBest[CPB][8];
  __shared__ int   sAssign[CPB][8]; // gt index assigned to pred slot, -1 = none
  __shared__ int   sMask[CPB][8];
  __shared__ float sAcc[CPB];

  const int tid = threadIdx.x;
  const float* gp = pred + (size_t)blockIdx.x * BLK_F;
  const float* gt = targ + (size_t)blockIdx.x * BLK_F;

  // ---- stage both slabs into LDS ----
#if HAVE_ASYNC_LDS
  for (int i = tid; i < BLK_V4; i += TPB) {
    __builtin_amdgcn_global_load_async_to_lds_b128(
        (gbl_v4i_p)(gp + 4 * i), (lds_v4i_p)(&sP[4 * i]), 0, 0);
    __builtin_amdgcn_global_load_async_to_lds_b128(
        (gbl_v4i_p)(gt + 4 * i), (lds_v4i_p)(&sT[4 * i]), 0, 0);
  }
#if HAVE_WAIT_ASYNC
  __builtin_amdgcn_s_wait_asynccnt(0);
#else
  asm volatile("s_wait_asynccnt 0" ::: "memory");
#endif
#else
  for (int i = tid; i < BLK_V4; i += TPB) {
    ((float4*)sP)[i] = ((const float4*)gp)[i];
    ((float4*)sT)[i] = ((const float4*)gt)[i];
  }
#endif
  __syncthreads();

  const int w = tid >> 5;   // wave id == cell-in-block
  const int l = tid & 31;   // lane id (wave32)
  const float* P = &sP[w * CELL_F];
  const float* T = &sT[w * CELL_F];

  const float AW[5] = {0.57273f, 1.87446f, 3.33843f, 7.88282f, 9.77052f};
  const float AH[5] = {0.677385f, 2.06253f, 5.47434f, 3.52778f, 9.16828f};

  // ---- phase A: decode boxes (lanes 0..4 = anchors) ----
  if (l < NA) {
    const int a = l;
    const float* pr = P + a * ND;
    const float* tr = T + a * ND;
    sBox[w][a * 4 + 0] = sigmoid_(pr[0]);
    sBox[w][a * 4 + 1] = sigmoid_(pr[1]);
    sBox[w][a * 4 + 2] = expf(pr[2]) * AW[a];
    sBox[w][a * 4 + 3] = expf(pr[3]) * AH[a];
    sBox[w][20 + a * 4 + 0] = tr[0];
    sBox[w][20 + a * 4 + 1] = tr[1];
    sBox[w][20 + a * 4 + 2] = expf(tr[2]) * AW[a];
    sBox[w][20 + a * 4 + 3] = expf(tr[3]) * AH[a];
    sMask[w][a] = (tr[4] > 0.5f) ? 1 : 0;
  }
  __syncthreads();

  // ---- phase B: 5x5 IoU matrix (lanes 0..24) ----
  if (l < 25) {
    const int ap = l % 5, ag = l / 5;
    const float bx = sBox[w][ap * 4 + 0], by = sBox[w][ap * 4 + 1];
    const float bw = sBox[w][ap * 4 + 2], bh = sBox[w][ap * 4 + 3];
    const float gx = sBox[w][20 + ag * 4 + 0], gy = sBox[w][20 + ag * 4 + 1];
    const float gw = sBox[w][20 + ag * 4 + 2], gh = sBox[w][20 + ag * 4 + 3];
    const float b1x1 = bx - bw * 0.5f, b1y1 = by - bh * 0.5f;
    const float b1x2 = bx + bw * 0.5f, b1y2 = by + bh * 0.5f;
    const float b2x1 = gx - gw * 0.5f, b2y1 = gy - gh * 0.5f;
    const float b2x2 = gx + gw * 0.5f, b2y2 = gy + gh * 0.5f;
    const float iw = fmaxf(fminf(b1x2, b2x2) - fmaxf(b1x1, b2x1), 0.f);
    const float ih = fmaxf(fminf(b1y2, b2y2) - fmaxf(b1y1, b2y1), 0.f);
    const float inter = iw * ih;
    const float uni = (b1x2 - b1x1) * (b1y2 - b1y1) +
                      (b2x2 - b2x1) * (b2y2 - b2y1) - inter + 1e-9f;
    sIou[w][ag * 5 + ap] = inter / uni;
  }
  __syncthreads();

  // ---- phase C: argmax over pred anchors per gt anchor (first-index ties) ----
  if (l < NA) {
    const int ag = l;
    float best = sIou[w][ag * 5 + 0];
    int bi = 0;
#pragma unroll
    for (int ap = 1; ap < NA; ++ap) {
      const float v = sIou[w][ag * 5 + ap];
      if (v > best) { best = v; bi = ap; }
    }
    sBest[w][ag] = bi;
  }
  __syncthreads();

  // ---- phase D: greedy assignment (matches reference sequential loop) ----
  if (l == 0) {
    int taken = 0;
    int asg[NA] = {-1, -1, -1, -1, -1};
#pragma unroll
    for (int ag = 0; ag < NA; ++ag) {
      const int ap = sBest[w][ag];
      if (sMask[w][ag] && !((taken >> ap) & 1)) {
        asg[ap] = ag;
        taken |= (1 << ap);
      }
    }
#pragma unroll
    for (int ap = 0; ap < NA; ++ap) sAssign[w][ap] = asg[ap];
  }
  __syncthreads();

  // ---- phase E: loss terms (wave cooperates per anchor) ----
  float acc = 0.f;
  for (int ap = 0; ap < NA; ++ap) {
    const int src = sAssign[w][ap];   // wave-uniform
    const float* pr = P + ap * ND;
    if (src >= 0) {
      // log-softmax over 80 pred class logits
      float m = -3.402823466e38f;
      for (int c = l; c < NC; c += 32) m = fmaxf(m, pr[5 + c]);
      for (int off = 16; off > 0; off >>= 1) m = fmaxf(m, __shfl_xor(m, off, 32));
      float s = 0.f;
      for (int c = l; c < NC; c += 32) s += expf(pr[5 + c] - m);
      for (int off = 16; off > 0; off >>= 1) s += __shfl_xor(s, off, 32);
      // label = argmax over assigned target class values (first-index ties)
      const float* tr = T + src * ND;
      float bv = -3.402823466e38f;
      int bc = NC;
      for (int c = l; c < NC; c += 32) {
        const float v = tr[5 + c];
        if (v > bv) { bv = v; bc = c; }
      }
      for (int off = 16; off > 0; off >>= 1) {
        const float ov = __shfl_xor(bv, off, 32);
        const int oc = __shfl_xor(bc, off, 32);
        if (ov > bv || (ov == bv && oc < bc)) { bv = ov; bc = oc; }
      }
      if (l == 0) {
        const float picked = pr[5 + bc] - m - logf(s);
        const float dx = sBox[w][ap * 4 + 0] - tr[0];
        const float dy = sBox[w][ap * 4 + 1] - tr[1];
        const float dw = pr[2] - tr[2];
        const float dh = pr[3] - tr[3];
        acc += 5.f * (dx * dx + dy * dy)    // L_XY
             + 5.f * (dw * dw + dh * dh)    // L_WH
             + softplus_(-pr[4])            // L_OBJ * 1.0
             - picked;                      // L_CLS * 1.0
      }
    } else if (l == 0) {
      acc += 0.5f * softplus_(pr[4]);       // L_NOOBJ
    }
  }
  if (l == 0) sAcc[w] = acc;
  __syncthreads();
  if (tid == 0) {
    float s = 0.f;
#pragma unroll
    for (int i = 0; i < CPB; ++i) s += sAcc[i];
    partials[blockIdx.x] = s;
  }
}

__global__ __launch_bounds__(256) void yolo_loss_reduce(
    const float* __restrict__ partials, float* __restrict__ out) {
  __shared__ float buf[256];
  float s = 0.f;
  for (int i = threadIdx.x; i < NBLK; i += 256) s += partials[i];
  buf[threadIdx.x] = s;
  __syncthreads();
  for (int off = 128; off > 0; off >>= 1) {
    if ((int)threadIdx.x < off) buf[threadIdx.x] += buf[threadIdx.x + off];
    __syncthreads();
  }
  if (threadIdx.x == 0) out[0] = buf[0] * (1.0f / (float)QN);
}

extern "C" void kernel_launch(void* const* d_in, const int* in_sizes, int n_in,
                              void* d_out, int out_size, void* d_ws, size_t ws_size,
                              hipStream_t stream) {
  const float* pred = (const float*)d_in[0];
  const float* targ = (const float*)d_in[1];
  float* out = (float*)d_out;
  float* partials = (float*)d_ws;   // NBLK floats = 23104 B of scratch
  yolo_loss_partial<<<NBLK, TPB, 0, stream>>>(pred, targ, partials);
  yolo_loss_reduce<<<1, 256, 0, stream>>>(partials, out);
}